// CrossVariableMixingHydra_78022375899457
// MI455X (gfx1250) — compile-verified
//
#include <hip/hip_runtime.h>
#include <hip/hip_bf16.h>
#include <math.h>

// CrossVariableMixingHydra — fused pipeline for MI455X (gfx1250, wave32, WMMA + TDM).
// B=64, L=2048, C=512, R=64. fp32 end-to-end (matches reference dtype); GEMMs use
// v_wmma_f32_16x16x4_f32. Workload is HBM-bound: ~0.86 GB min traffic -> ~37us at
// 23.3 TB/s, so fp32 WMMA at K=4 is not the bottleneck. K1 streams x (268 MB)
// through the Tensor Data Mover into double-buffered LDS tiles (TENSORcnt overlap).
// All intermediates stored transposed (B,R,C) so every WMMA fragment access and
// every store is lane-coalesced.

#define BB 64
#define LL 2048
#define CC 512
#define RR 64

typedef float v2f __attribute__((ext_vector_type(2)));
typedef float v8f __attribute__((ext_vector_type(8)));
typedef unsigned int u32x4 __attribute__((ext_vector_type(4)));
typedef int i32x4 __attribute__((ext_vector_type(4)));
typedef int i32x8 __attribute__((ext_vector_type(8)));

// ---------------------------------------------------------------------------
// TDM: 2D tile (tile_d0 x tile_d1 elements, 4B each) global -> LDS.
// D# group0: count=1 | lds_addr | global_addr[56:0] | type=2
// D# group1: data_size=4B | tensor dims | tile dims | dim0 stride
// groups 2/3 zero (<=2D tensor). Tracked by TENSORcnt.
// ---------------------------------------------------------------------------
__device__ __forceinline__ void tdm_load_2d(void* lds_dst, const void* gsrc,
                                            unsigned tensor_d0, unsigned tensor_d1,
                                            unsigned stride_d0,
                                            unsigned tile_d0, unsigned tile_d1) {
  unsigned long long ga = (unsigned long long)(size_t)gsrc;
  u32x4 g0;
  g0[0] = 1u;                                   // count=1 (valid user descriptor)
  g0[1] = (unsigned)(size_t)lds_dst;            // LDS byte address (low 32 flat bits)
  g0[2] = (unsigned)ga;                         // global_addr[31:0]
  g0[3] = (unsigned)((ga >> 32) & 0x1FFFFFFu) | (2u << 30);  // addr[56:32], type=2
  i32x8 g1;
  g1[0] = (int)(2u << 16);                                      // data_size=4B
  g1[1] = (int)((tensor_d0 & 0xFFFFu) << 16);                   // tensor_dim0 lo16
  g1[2] = (int)(((tensor_d0 >> 16) & 0xFFFFu) | ((tensor_d1 & 0xFFFFu) << 16));
  g1[3] = (int)(((tensor_d1 >> 16) & 0xFFFFu) | (tile_d0 << 16));  // tile_dim0
  g1[4] = (int)(tile_d1 & 0xFFFFu);                             // tile_dim1, dim2=0
  g1[5] = (int)stride_d0;                                       // dim0_stride lo32
  g1[6] = 0;
  g1[7] = 0;
  i32x4 z4 = {0, 0, 0, 0};
#if __clang_major__ >= 23
  i32x8 z8 = {0, 0, 0, 0, 0, 0, 0, 0};
  __builtin_amdgcn_tensor_load_to_lds(g0, g1, z4, z4, z8, 0);
#else
  __builtin_amdgcn_tensor_load_to_lds(g0, g1, z4, z4, 0);
#endif
}

// ---------------------------------------------------------------------------
// K0: zero the global-feature accumulator (atomic target) every launch.
// ---------------------------------------------------------------------------
__global__ void k0_zero(float* __restrict__ gf, int n) {
  int i = blockIdx.x * blockDim.x + threadIdx.x;
  if (i < n) gf[i] = 0.0f;
}

// ---------------------------------------------------------------------------
// K1: hlowT[b,r,c] = sum_l Wd[r,l] * x[b,l,c] + bd[r]     (D = Wd @ x[b])
// grid (C/64, B), block 128 (4 waves). Wave w: M-tile r0=w*16, N = 64 c's.
// A = Wd rows (contiguous v2f loads). B = x tile, DMA'd by TDM into
// double-buffered LDS (128 l x 64 c = 32 KB per buffer), issued by wave 0,
// overlapped with compute via TENSORcnt.
// ---------------------------------------------------------------------------
__global__ __launch_bounds__(128) void k1_down(const float* __restrict__ x,
                                               const float* __restrict__ Wd,
                                               const float* __restrict__ bd,
                                               float* __restrict__ hlowT) {
  __shared__ float sX[2][128 * 64];  // [buf][kLocal*64 + c]
  const int b    = blockIdx.y;
  const int c0   = blockIdx.x * 64;
  const int wv   = threadIdx.x >> 5;
  const int lane = threadIdx.x & 31;
  const int m    = lane & 15;
  const int koff = (lane >> 4) * 2;
  const int r0   = wv * 16;

  const float* xb = x + (size_t)b * LL * CC + c0;  // tile column origin
  v8f acc[4] = {};

  if (threadIdx.x < 32)  // wave 0 issues the DMAs (EXEC ignored by TDM)
    tdm_load_2d(&sX[0][0], xb, 64u, 128u, (unsigned)CC, 64u, 128u);

  for (int ch = 0; ch < 16; ++ch) {
    if (threadIdx.x < 32) {
      if (ch + 1 < 16)
        tdm_load_2d(&sX[(ch + 1) & 1][0], xb + (size_t)(ch + 1) * 128 * CC,
                    64u, 128u, (unsigned)CC, 64u, 128u);
      if (ch + 1 < 16)
        __builtin_amdgcn_s_wait_tensorcnt(1);  // chunk ch landed (in-order TDM)
      else
        __builtin_amdgcn_s_wait_tensorcnt(0);
    }
    __syncthreads();  // publish buffer ch&1 to all 4 waves

    const float* sx = sX[ch & 1];
    const float* wd = Wd + (size_t)(r0 + m) * LL + (size_t)ch * 128 + koff;
    for (int kk = 0; kk < 128; kk += 4) {
      v2f a = *(const v2f*)(wd + kk);  // A 16x4: contiguous k-pair per lane
#pragma unroll
      for (int t = 0; t < 4; ++t) {
        v2f bf;  // B 4x16: row k striped over lanes (c contiguous in LDS)
        bf.x = sx[(kk + koff) * 64 + t * 16 + m];
        bf.y = sx[(kk + koff + 1) * 64 + t * 16 + m];
        acc[t] = __builtin_amdgcn_wmma_f32_16x16x4_f32(
            false, a, false, bf, (short)0, acc[t], false, false);
      }
    }
    __syncthreads();  // done reading buffer before it is re-filled
  }

  // D tile: M=r, N=c -> (B,R,C) stores are lane-coalesced.
  const int hi = lane >> 4;
  float* ho = hlowT + ((size_t)b * RR + r0) * CC + c0;
#pragma unroll
  for (int t = 0; t < 4; ++t)
#pragma unroll
    for (int v = 0; v < 8; ++v) {
      int rr = v + 8 * hi;
      ho[(size_t)rr * CC + t * 16 + m] = acc[t][v] + bd[r0 + rr];
    }
}

// ---------------------------------------------------------------------------
// K2a: Q/K/V/G = h_low @ W{q,k,v,g}^T, l2norm(Q), l2norm(K), gate=sigmoid(G+bg),
//      gf[b,:] += sum_c Kn*V   (global atomicAdd, gf zeroed by K0)
// grid (C/64, B), block 128. A-fragments read hlowT (B,R,C) -> coalesced.
// All four 64x64 weights staged transposed in LDS (64 KB), shared A-fragment.
// ---------------------------------------------------------------------------
__global__ __launch_bounds__(128) void k2a_qkvg(
    const float* __restrict__ hlowT, const float* __restrict__ Wq,
    const float* __restrict__ Wk, const float* __restrict__ Wv,
    const float* __restrict__ Wg, const float* __restrict__ bg,
    float* __restrict__ QnT, float* __restrict__ gateT, float* __restrict__ gf) {
  __shared__ float sW[4][64 * 64];  // sW[w][k*64+s] = W[s][k]
  const int b    = blockIdx.y;
  const int c0w  = blockIdx.x * 64 + (threadIdx.x >> 5) * 16;
  const int lane = threadIdx.x & 31;
  const int n16  = lane & 15;
  const int koff = (lane >> 4) * 2;

  const float* Ws[4] = {Wq, Wk, Wv, Wg};
#pragma unroll
  for (int w = 0; w < 4; ++w)
    for (int i = threadIdx.x; i < 64 * 64; i += 128) {
      int s = i >> 6, r = i & 63;
      sW[w][r * 64 + s] = Ws[w][i];  // transpose on store
    }
  __syncthreads();

  v8f acc[4][4] = {};  // [matrix][ntile]
  const float* hT = hlowT + (size_t)b * RR * CC + c0w;
  for (int kk = 0; kk < RR; kk += 4) {
    v2f a;  // A row m = channel c0w+n16, k = r  -> coalesced over lanes
    a.x = hT[(size_t)(kk + koff) * CC + n16];
    a.y = hT[(size_t)(kk + koff + 1) * CC + n16];
#pragma unroll
    for (int w = 0; w < 4; ++w)
#pragma unroll
      for (int t = 0; t < 4; ++t) {
        v2f bf;
        bf.x = sW[w][(kk + koff) * 64 + t * 16 + n16];
        bf.y = sW[w][(kk + koff + 1) * 64 + t * 16 + n16];
        acc[w][t] = __builtin_amdgcn_wmma_f32_16x16x4_f32(
            false, a, false, bf, (short)0, acc[w][t], false, false);
      }
  }

  const int hi = lane >> 4;
  // l2-normalize Q (acc[0]) and K (acc[1]) channel rows: half-wave xor shuffles.
#pragma unroll
  for (int w = 0; w < 2; ++w)
#pragma unroll
    for (int v = 0; v < 8; ++v) {
      float ss = 0.0f;
#pragma unroll
      for (int t = 0; t < 4; ++t) ss += acc[w][t][v] * acc[w][t][v];
      ss += __shfl_xor(ss, 1, 32);
      ss += __shfl_xor(ss, 2, 32);
      ss += __shfl_xor(ss, 4, 32);
      ss += __shfl_xor(ss, 8, 32);
      float rn = 1.0f / fmaxf(sqrtf(ss), 1e-12f);
#pragma unroll
      for (int t = 0; t < 4; ++t) acc[w][t][v] *= rn;
    }
  // gate = sigmoid(G + bg)
#pragma unroll
  for (int t = 0; t < 4; ++t) {
    float bgv = bg[t * 16 + n16];
#pragma unroll
    for (int v = 0; v < 8; ++v) {
      float g = acc[3][t][v] + bgv;
      acc[3][t][v] = 1.0f / (1.0f + __expf(-g));
    }
  }
  // gf[b,r] += sum over this wave's 16 channels of Kn*V
#pragma unroll
  for (int t = 0; t < 4; ++t) {
    float s = 0.0f;
#pragma unroll
    for (int v = 0; v < 8; ++v) s += acc[1][t][v] * acc[2][t][v];
    s += __shfl_xor(s, 16, 32);  // combine lane-halves (rows m and m+8)
    if (lane < 16) atomicAdd(&gf[b * RR + t * 16 + n16], s);
  }
  // store QnT, gateT in (B,R,C): 8 consecutive floats per lane per tile
  float* qb = QnT + (size_t)b * RR * CC + c0w;
  float* gb = gateT + (size_t)b * RR * CC + c0w;
#pragma unroll
  for (int t = 0; t < 4; ++t)
#pragma unroll
    for (int v = 0; v < 8; ++v) {
      size_t idx = (size_t)(t * 16 + n16) * CC + v + 8 * hi;
      qb[idx] = acc[0][t][v];
      gb[idx] = acc[3][t][v];
    }
}

// ---------------------------------------------------------------------------
// K2b: h_attnT = QnT * gf[b,r] * gateT   ((B,R,C) layout; writes h_low buffer)
// ---------------------------------------------------------------------------
__global__ void k2b_attn(const float* __restrict__ QnT,
                         const float* __restrict__ gateT,
                         const float* __restrict__ gf,
                         float* __restrict__ hattnT, int total) {
  int i = blockIdx.x * blockDim.x + threadIdx.x;
  if (i < total) {
    int r = (i >> 9) & (RR - 1);   // (B,R,C): c fastest (512), then r (64)
    int b = i >> 15;
    hattnT[i] = QnT[i] * gf[b * RR + r] * gateT[i];
  }
}

// ---------------------------------------------------------------------------
// K3: out[b,l,c] = LN_c( x[b,l,c] + alpha*(h_attn[b,c,:]·Wu[l,:] + bu[l]) )
// grid (L/16, B), block 256 (8 waves). WG owns ALL 512 channels of a 16-wide
// l-tile; LayerNorm reduction is shfl + one tiny LDS pass (saves a 536 MB
// global round-trip). A-fragments read hattnT (B,R,C) -> coalesced.
// ---------------------------------------------------------------------------
__global__ __launch_bounds__(256) void k3_up_ln(
    const float* __restrict__ hattnT, const float* __restrict__ Wu,
    const float* __restrict__ bu, const float* __restrict__ x,
    const float* __restrict__ gamma, const float* __restrict__ beta,
    const float* __restrict__ alphaP, float* __restrict__ out) {
  __shared__ float sWu[64 * 16];  // [k][n], conflict-free B-frag reads
  __shared__ float sSum[8][16];
  __shared__ float sSsq[8][16];
  const int b    = blockIdx.y;
  const int l0   = blockIdx.x * 16;
  const int w    = threadIdx.x >> 5;
  const int lane = threadIdx.x & 31;
  const int n16  = lane & 15;
  const int koff = (lane >> 4) * 2;

  for (int i = threadIdx.x; i < 16 * 64; i += 256) {
    int n = i >> 6, k = i & 63;
    sWu[k * 16 + n] = Wu[(size_t)(l0 + n) * RR + k];
  }
  __syncthreads();

  const int cbase = w * 64;
  const float* hT = hattnT + (size_t)b * RR * CC + cbase;
  v8f acc[4] = {};
  for (int kk = 0; kk < RR; kk += 4) {
    v2f bf;
    bf.x = sWu[(kk + koff) * 16 + n16];
    bf.y = sWu[(kk + koff + 1) * 16 + n16];
#pragma unroll
    for (int t = 0; t < 4; ++t) {
      v2f a;  // A row m = channel cbase+t*16+n16, k = r -> coalesced
      a.x = hT[(size_t)(kk + koff) * CC + t * 16 + n16];
      a.y = hT[(size_t)(kk + koff + 1) * CC + t * 16 + n16];
      acc[t] = __builtin_amdgcn_wmma_f32_16x16x4_f32(
          false, a, false, bf, (short)0, acc[t], false, false);
    }
  }

  // residual + bias + alpha, and LN partial sums
  const int hi  = lane >> 4;
  const int l   = l0 + n16;  // this lane's output row (fixed)
  const float al  = alphaP[0];
  const float buv = bu[l];
  const float* xrow = x + ((size_t)b * LL + l) * CC;
  float s = 0.0f, s2 = 0.0f;
#pragma unroll
  for (int t = 0; t < 4; ++t)
#pragma unroll
    for (int v = 0; v < 8; ++v) {
      int c = cbase + t * 16 + v + 8 * hi;  // consecutive c per vgpr index
      float y = xrow[c] + al * (acc[t][v] + buv);
      acc[t][v] = y;
      s += y;
      s2 += y * y;
    }
  s  += __shfl_xor(s, 16, 32);  // other lane-half holds the other 32 channels
  s2 += __shfl_xor(s2, 16, 32);
  if (lane < 16) { sSum[w][n16] = s; sSsq[w][n16] = s2; }
  __syncthreads();
  float tot = 0.0f, tot2 = 0.0f;
#pragma unroll
  for (int ww = 0; ww < 8; ++ww) { tot += sSum[ww][n16]; tot2 += sSsq[ww][n16]; }
  const float mu   = tot * (1.0f / CC);
  const float var  = tot2 * (1.0f / CC) - mu * mu;
  const float rstd = rsqrtf(var + 1e-5f);

  float* orow = out + ((size_t)b * LL + l) * CC;
#pragma unroll
  for (int t = 0; t < 4; ++t)
#pragma unroll
    for (int v = 0; v < 8; ++v) {
      int c = cbase + t * 16 + v + 8 * hi;
      orow[c] = (acc[t][v] - mu) * rstd * gamma[c] + beta[c];
    }
}

// ---------------------------------------------------------------------------
extern "C" void kernel_launch(void* const* d_in, const int* in_sizes, int n_in,
                              void* d_out, int out_size, void* d_ws,
                              size_t ws_size, hipStream_t stream) {
  const float* x     = (const float*)d_in[0];
  const float* Wd    = (const float*)d_in[1];
  const float* bd    = (const float*)d_in[2];
  const float* Wq    = (const float*)d_in[3];
  const float* Wk    = (const float*)d_in[4];
  const float* Wv    = (const float*)d_in[5];
  const float* Wg    = (const float*)d_in[6];
  const float* bg    = (const float*)d_in[7];
  const float* Wu    = (const float*)d_in[8];
  const float* bu    = (const float*)d_in[9];
  const float* gamma = (const float*)d_in[10];
  const float* beta  = (const float*)d_in[11];
  const float* alpha = (const float*)d_in[12];
  float* out = (float*)d_out;

  // workspace layout (floats): hlowT(2M) | QnT(2M) | gateT(2M) | gf(4K)
  float* ws    = (float*)d_ws;
  float* hlowT = ws;                              // (B,R,C); reused as h_attnT
  float* QnT   = ws + (size_t)BB * CC * RR;
  float* gateT = ws + 2 * (size_t)BB * CC * RR;
  float* gf    = ws + 3 * (size_t)BB * CC * RR;   // (B,R)

  const int nGf = BB * RR;
  k0_zero<<<(nGf + 255) / 256, 256, 0, stream>>>(gf, nGf);

  k1_down<<<dim3(CC / 64, BB), 128, 0, stream>>>(x, Wd, bd, hlowT);

  k2a_qkvg<<<dim3(CC / 64, BB), 128, 0, stream>>>(hlowT, Wq, Wk, Wv, Wg, bg,
                                                  QnT, gateT, gf);

  const int total = BB * CC * RR;
  k2b_attn<<<(total + 255) / 256, 256, 0, stream>>>(QnT, gateT, gf, hlowT, total);

  k3_up_ln<<<dim3(LL / 16, BB), 256, 0, stream>>>(hlowT, Wu, bu, x, gamma, beta,
                                                  alpha, out);
}